// MeshDepthFitterEnergy_33019708572398
// MI455X (gfx1250) — compile-verified
//
#include <hip/hip_runtime.h>
#include <hip/hip_bf16.h>
#include <math.h>

#define MAXD      6.0f
#define CREGU_C   2000.0f
#define IMG_H     240
#define IMG_W     320
#define FOCAL     (2.0f * (float)IMG_W)
#define TILE_CH   64          // face-chunks (of 4 faces) per LDS tile

typedef __attribute__((ext_vector_type(2))) float v2f;
typedef __attribute__((ext_vector_type(4))) float v4f;
typedef __attribute__((ext_vector_type(8))) float v8f;
typedef __attribute__((ext_vector_type(4))) int   v4i;

// ---------------------------------------------------------------------------
// gfx1250 async global->LDS helpers (guarded; plain-copy fallback)
// ---------------------------------------------------------------------------
__device__ __forceinline__ void asyncCopy16(const float* g, float* l) {
#if __has_builtin(__builtin_amdgcn_global_load_async_to_lds_b128)
    typedef __attribute__((address_space(1))) v4i* gptr_t;
    typedef __attribute__((address_space(3))) v4i* lptr_t;
    __builtin_amdgcn_global_load_async_to_lds_b128((gptr_t)g, (lptr_t)l, 0, 0);
#else
    *(v4f*)l = *(const v4f*)g;
#endif
}

__device__ __forceinline__ void asyncWait() {
#if __has_builtin(__builtin_amdgcn_s_wait_asynccnt)
    __builtin_amdgcn_s_wait_asynccnt(0);
#else
    asm volatile("s_wait_asynccnt 0" ::: "memory");
#endif
}

// ---------------------------------------------------------------------------
// Pass 0: zero accumulators + Laplacian scratch
// ---------------------------------------------------------------------------
__global__ void zero_kernel(float* p, int n) {
    int i = blockIdx.x * blockDim.x + threadIdx.x;
    if (i < n) p[i] = 0.0f;
}

// ---------------------------------------------------------------------------
// Pass 1: reductions (vertex mean, vinit mean/std) + vertex transform/project
// ---------------------------------------------------------------------------
__global__ __launch_bounds__(256) void prep_kernel(
    const float* __restrict__ verts, const float* __restrict__ quat,
    const float* __restrict__ trans, const float* __restrict__ vinit,
    float* __restrict__ xy, float* __restrict__ zcv, int nv)
{
    __shared__ float red[256];
    __shared__ float tot[9];

    float s[9];
#pragma unroll
    for (int k = 0; k < 9; ++k) s[k] = 0.0f;

    for (int i = threadIdx.x; i < nv; i += blockDim.x) {
        float vx = verts[i*3+0], vy = verts[i*3+1], vz = verts[i*3+2];
        float ix = vinit[i*3+0], iy = vinit[i*3+1], iz = vinit[i*3+2];
        s[0] += vx; s[1] += vy; s[2] += vz;
        s[3] += ix; s[4] += iy; s[5] += iz;
        s[6] += ix*ix; s[7] += iy*iy; s[8] += iz*iz;
    }
#pragma unroll
    for (int k = 0; k < 9; ++k) {
        red[threadIdx.x] = s[k];
        __syncthreads();
        for (int st = 128; st > 0; st >>= 1) {
            if ((int)threadIdx.x < st) red[threadIdx.x] += red[threadIdx.x + st];
            __syncthreads();
        }
        if (threadIdx.x == 0) tot[k] = red[0];
        __syncthreads();
    }

    const float invn = 1.0f / (float)nv;
    const float mvx = tot[0]*invn, mvy = tot[1]*invn, mvz = tot[2]*invn;
    const float cix = tot[3]*invn, ciy = tot[4]*invn, ciz = tot[5]*invn;
    float sdx = sqrtf(fmaxf(tot[6]*invn - cix*cix, 0.0f));
    float sdy = sqrtf(fmaxf(tot[7]*invn - ciy*ciy, 0.0f));
    float sdz = sqrtf(fmaxf(tot[8]*invn - ciz*ciz, 0.0f));
    const float radius = fmaxf(sdx, fmaxf(sdy, sdz));

    const float camx = cix - 0.5f * radius;
    const float camy = ciy;
    const float camz = ciz + 5.0f * radius;

    float qx = quat[0], qy = quat[1], qz = quat[2], qw = quat[3];
    float qn = rsqrtf(qx*qx + qy*qy + qz*qz + qw*qw);
    qx *= qn; qy *= qn; qz *= qn; qw *= qn;
    const float tx = trans[0], ty = trans[1], tz = trans[2];

    for (int i = threadIdx.x; i < nv; i += blockDim.x) {
        float vx = verts[i*3+0] - mvx;
        float vy = verts[i*3+1] - mvy;
        float vz = verts[i*3+2] - mvz;
        float uvx = qy*vz - qz*vy;
        float uvy = qz*vx - qx*vz;
        float uvz = qx*vy - qy*vx;
        float uux = qy*uvz - qz*uvy;
        float uuy = qz*uvx - qx*uvz;
        float uuz = qx*uvy - qy*uvx;
        float vtx = vx + 2.0f*(qw*uvx + uux) + tx;
        float vty = vy + 2.0f*(qw*uvy + uuy) + ty;
        float vtz = vz + 2.0f*(qw*uvz + uuz) + tz;
        float pcx = vtx - camx;
        float pcy = -vty + camy;
        float pcz = -vtz + camz;
        float u = FOCAL * pcx / pcz + (float)IMG_W * 0.5f;
        float v = FOCAL * pcy / pcz + (float)IMG_H * 0.5f;
        xy[i*2+0] = u;
        xy[i*2+1] = v;
        zcv[i] = pcz;
    }
}

// ---------------------------------------------------------------------------
// Pass 2: per-face affine edge coefficients, meta, bbox
// ---------------------------------------------------------------------------
__global__ void coef_kernel(
    const float* __restrict__ xy, const float* __restrict__ zcv,
    const int* __restrict__ faces,
    float* __restrict__ coef, float* __restrict__ meta,
    float* __restrict__ fbox, int nf, int nfPad)
{
    int f = blockIdx.x * blockDim.x + threadIdx.x;
    if (f >= nfPad) return;
    float* cp = coef + (size_t)f * 16;
    if (f >= nf) {
#pragma unroll
        for (int k = 0; k < 16; ++k) cp[k] = 0.0f;
        meta[f*2+0] = 0.0f; meta[f*2+1] = 0.0f;
        fbox[f*4+0] =  1e30f; fbox[f*4+1] =  1e30f;
        fbox[f*4+2] = -1e30f; fbox[f*4+3] = -1e30f;
        return;
    }
    int ia = faces[f*3+0], ib = faces[f*3+1], ic = faces[f*3+2];
    float ax = xy[ia*2+0], ay = xy[ia*2+1];
    float bx = xy[ib*2+0], by = xy[ib*2+1];
    float cx = xy[ic*2+0], cy = xy[ic*2+1];
    float za = zcv[ia], zb = zcv[ib], zc2 = zcv[ic];

    float Aa = by - cy, Ba = cx - bx, Ca = (cy - by)*bx - (cx - bx)*by; // b->c
    float Ab = cy - ay, Bb = ax - cx, Cb = (ay - cy)*cx - (ax - cx)*cy; // c->a
    float Ac = ay - by, Bc = bx - ax, Cc = (by - ay)*ax - (bx - ax)*ay; // a->b
    float area = Ca + Cb + Cc;
    bool valid = fabsf(area) > 1e-9f;
    float sgn = valid ? (area > 0.0f ? 1.0f : -1.0f) : 0.0f;
    float rcp = valid ? 1.0f / area : 0.0f;
    float An = Aa*za + Ab*zb + Ac*zc2;
    float Bn = Ba*za + Bb*zb + Bc*zc2;
    float Cn = Ca*za + Cb*zb + Cc*zc2;

    cp[0]  = Aa; cp[1]  = Ba; cp[2]  = Ca; cp[3]  = 0.0f;
    cp[4]  = Ab; cp[5]  = Bb; cp[6]  = Cb; cp[7]  = 0.0f;
    cp[8]  = Ac; cp[9]  = Bc; cp[10] = Cc; cp[11] = 0.0f;
    cp[12] = An; cp[13] = Bn; cp[14] = Cn; cp[15] = 0.0f;
    meta[f*2+0] = sgn;
    meta[f*2+1] = rcp;
    fbox[f*4+0] = fminf(ax, fminf(bx, cx));
    fbox[f*4+1] = fminf(ay, fminf(by, cy));
    fbox[f*4+2] = fmaxf(ax, fmaxf(bx, cx));
    fbox[f*4+3] = fmaxf(ay, fmaxf(by, cy));
}

// ---------------------------------------------------------------------------
// Pass 3: merge face bboxes into per-chunk (4 faces) bboxes
// ---------------------------------------------------------------------------
__global__ void cbox_kernel(const float* __restrict__ fbox,
                            float* __restrict__ cbox, int nChunks)
{
    int c = blockIdx.x * blockDim.x + threadIdx.x;
    if (c >= nChunks) return;
    float x0 = 1e30f, y0 = 1e30f, x1 = -1e30f, y1 = -1e30f;
#pragma unroll
    for (int k = 0; k < 4; ++k) {
        const float* b = fbox + (size_t)(c*4+k)*4;
        x0 = fminf(x0, b[0]); y0 = fminf(y0, b[1]);
        x1 = fmaxf(x1, b[2]); y1 = fmaxf(y1, b[3]);
    }
    cbox[c*4+0] = x0; cbox[c*4+1] = y0; cbox[c*4+2] = x1; cbox[c*4+3] = y1;
}

// ---------------------------------------------------------------------------
// Pass 4: WMMA rasterizer, async double-buffered LDS staging of the face
// tables, branchless z-buffer epilogue, data-energy accumulation.
//   A [16x4] rows = (face<<2)|quantity, K coeffs = (A,B,C,0)   (from LDS)
//   B [4x16]  rows = (PX, 1, PY, 0), cols = pixels
//   D [16x16] lane n -> pixel n%16; lanes 0-15 -> faces 0,1; 16-31 -> 2,3
// ---------------------------------------------------------------------------
__global__ __launch_bounds__(256) void raster_kernel(
    const float* __restrict__ coef, const float* __restrict__ meta,
    const float* __restrict__ cbox, const float* __restrict__ img,
    float* __restrict__ acc, int nChunksPad)
{
    __shared__ __align__(16) float coefLds[2][TILE_CH * 64]; // 16 KB / buf
    __shared__ __align__(16) float metaLds[2][TILE_CH * 8];  //  2 KB / buf
    __shared__ __align__(16) float cboxLds[2][TILE_CH * 4];  //  1 KB / buf

    const int tid  = threadIdx.x;
    const int lane = tid & 31;
    const int wave = tid >> 5;
    const int x0 = blockIdx.x * 16;
    const int y  = blockIdx.y * 8 + wave;

    const float PX = (float)(x0 + (lane & 15)) + 0.5f;
    const float PY = (float)y + 0.5f;

    v2f bmat;
    bmat.x = (lane < 16) ? PX : PY;      // K0 = PX, K2 = PY
    bmat.y = (lane < 16) ? 1.0f : 0.0f;  // K1 = 1,  K3 = 0

    const int m    = lane & 15;
    const int qq   = m & 3;
    const int fo   = m >> 2;
    const int kb   = (lane < 16) ? 0 : 2;
    const int fsel = (lane < 16) ? 0 : 2;

    const int nTiles = nChunksPad / TILE_CH;

    // issue async fill of tile 0 into buffer 0
    {
        const float* gc = coef;
        const float* gm = meta;
        const float* gb = cbox;
#pragma unroll
        for (int i = 0; i < 4; ++i) {           // 1024 x b128 coef
            int k = tid + i * 256;
            asyncCopy16(gc + (size_t)k * 4, &coefLds[0][k * 4]);
        }
        if (tid < 128) asyncCopy16(gm + (size_t)tid * 4, &metaLds[0][tid * 4]);
        if (tid < 64)  asyncCopy16(gb + (size_t)tid * 4, &cboxLds[0][tid * 4]);
    }
    asyncWait();
    __syncthreads();

    float zmin = MAXD;
    int cur = 0;

    for (int t = 0; t < nTiles; ++t) {
        // prefetch next tile into the other buffer (async, no wait yet)
        if (t + 1 < nTiles) {
            const int nb2 = cur ^ 1;
            const float* gc = coef + (size_t)(t + 1) * TILE_CH * 64;
            const float* gm = meta + (size_t)(t + 1) * TILE_CH * 8;
            const float* gb = cbox + (size_t)(t + 1) * TILE_CH * 4;
#pragma unroll
            for (int i = 0; i < 4; ++i) {
                int k = tid + i * 256;
                asyncCopy16(gc + (size_t)k * 4, &coefLds[nb2][k * 4]);
            }
            if (tid < 128) asyncCopy16(gm + (size_t)tid * 4, &metaLds[nb2][tid * 4]);
            if (tid < 64)  asyncCopy16(gb + (size_t)tid * 4, &cboxLds[nb2][tid * 4]);
        }

        // process current tile from LDS
        for (int cc = 0; cc < TILE_CH; ++cc) {
            float bx0 = cboxLds[cur][cc*4+0], by0 = cboxLds[cur][cc*4+1];
            float bx1 = cboxLds[cur][cc*4+2], by1 = cboxLds[cur][cc*4+3];
            bool ov = (bx1 >= (float)x0 - 1.0f) && (bx0 <= (float)(x0 + 16) + 1.0f) &&
                      (by1 >= PY - 1.0f) && (by0 <= PY + 1.0f);
            if (!__builtin_amdgcn_readfirstlane((int)ov)) continue;

            // A-matrix rows from LDS (ds_load_b64 per lane)
            v2f amat = *(const v2f*)&coefLds[cur][(cc * 4 + fo) * 16 + qq * 4 + kb];

            v8f czero = {};
            v8f d = __builtin_amdgcn_wmma_f32_16x16x4_f32(
                false, amat, false, bmat, (short)0, czero, false, false);

            // meta for this lane's two faces: 16B-aligned LDS vector load
            v4f mm = *(const v4f*)&metaLds[cur][(cc * 4 + fsel) * 2];
            float s0 = mm.x, r0 = mm.y, s1 = mm.z, r1 = mm.w;

            // branchless candidate min (keeps EXEC untouched between WMMAs)
            float zi0 = d[3] * r0;
            bool ok0 = (d[0]*s0 >= 0.0f) & (d[1]*s0 >= 0.0f) &
                       (d[2]*s0 >= 0.0f) & (zi0 > 0.0f);
            zmin = fminf(zmin, ok0 ? zi0 : MAXD);

            float zi1 = d[7] * r1;
            bool ok1 = (d[4]*s1 >= 0.0f) & (d[5]*s1 >= 0.0f) &
                       (d[6]*s1 >= 0.0f) & (zi1 > 0.0f);
            zmin = fminf(zmin, ok1 ? zi1 : MAXD);
        }

        asyncWait();         // next tile's async fills are complete
        __syncthreads();     // all waves done reading current buffer
        cur ^= 1;
    }

    // merge the two face-halves covering the same pixel
    zmin = fminf(zmin, __shfl_xor(zmin, 16, 32));

    float err2 = 0.0f;
    if (lane < 16 && y < IMG_H) {
        float depth = fminf(fmaxf(zmin, 0.0f), MAXD);
        float e = depth - img[y * IMG_W + x0 + lane];
        err2 = e * e;
    }
#pragma unroll
    for (int off = 16; off > 0; off >>= 1)
        err2 += __shfl_down(err2, off, 32);
    if (lane == 0) atomicAdd(&acc[0], err2);
}

// ---------------------------------------------------------------------------
// Pass 5: edge scatter for graph Laplacian
// ---------------------------------------------------------------------------
__global__ void edge_kernel(const float* __restrict__ verts,
                            const float* __restrict__ vinit,
                            const int* __restrict__ edges,
                            float* __restrict__ deg, float* __restrict__ nb, int ne)
{
    int e = blockIdx.x * blockDim.x + threadIdx.x;
    if (e >= ne) return;
    int i = edges[e*2+0], j = edges[e*2+1];
    atomicAdd(&deg[i], 1.0f);
    atomicAdd(&deg[j], 1.0f);
#pragma unroll
    for (int k = 0; k < 3; ++k) {
        float di = verts[i*3+k] - vinit[i*3+k];
        float dj = verts[j*3+k] - vinit[j*3+k];
        atomicAdd(&nb[i*3+k], dj);
        atomicAdd(&nb[j*3+k], di);
    }
}

// ---------------------------------------------------------------------------
// Pass 6: per-vertex Laplacian energy
// ---------------------------------------------------------------------------
__global__ void vertex_kernel(const float* __restrict__ verts,
                              const float* __restrict__ vinit,
                              const float* __restrict__ deg,
                              const float* __restrict__ nb,
                              float* __restrict__ acc, int nv)
{
    int i = blockIdx.x * blockDim.x + threadIdx.x;
    if (i >= nv) return;
    float dg = deg[i], sum = 0.0f;
#pragma unroll
    for (int k = 0; k < 3; ++k) {
        float d  = verts[i*3+k] - vinit[i*3+k];
        float ld = dg * d - nb[i*3+k];
        sum += ld * ld;
    }
    atomicAdd(&acc[1], sum);
}

// ---------------------------------------------------------------------------
// Pass 7: combine
// ---------------------------------------------------------------------------
__global__ void final_kernel(const float* __restrict__ acc, float* __restrict__ out)
{
    out[0] = acc[0] + 0.5f * CREGU_C * acc[1];
}

// ---------------------------------------------------------------------------
extern "C" void kernel_launch(void* const* d_in, const int* in_sizes, int n_in,
                              void* d_out, int out_size, void* d_ws, size_t ws_size,
                              hipStream_t stream)
{
    const float* verts = (const float*)d_in[0];
    const float* quat  = (const float*)d_in[1];
    const float* trans = (const float*)d_in[2];
    const float* vinit = (const float*)d_in[3];
    const float* img   = (const float*)d_in[4];
    const int*   faces = (const int*)d_in[5];
    const int*   edges = (const int*)d_in[6];
    float*       out   = (float*)d_out;

    const int nv = in_sizes[0] / 3;
    const int nf = in_sizes[5] / 3;
    const int ne = in_sizes[6] / 2;
    const int nChunks    = (nf + 3) / 4;
    const int nTiles     = (nChunks + TILE_CH - 1) / TILE_CH;
    const int nChunksPad = nTiles * TILE_CH;
    const int nfPad      = nChunksPad * 4;

    float* ws = (float*)d_ws;
    size_t off = 0;
    float* xy   = ws + off; off += (size_t)nv * 2;
    float* zcv  = ws + off; off += (size_t)nv;
    float* coef = ws + off; off += (size_t)nfPad * 16;
    float* meta = ws + off; off += (size_t)nfPad * 2;
    float* fbox = ws + off; off += (size_t)nfPad * 4;
    float* cbox = ws + off; off += (size_t)nChunksPad * 4;
    float* deg  = ws + off; off += (size_t)nv;
    float* nb   = ws + off; off += (size_t)nv * 3;
    float* acc  = ws + off; off += 2;

    {
        int nz = nv + nv * 3 + 2;  // deg + nb + acc (contiguous)
        zero_kernel<<<(nz + 255) / 256, 256, 0, stream>>>(deg, nz);
    }
    prep_kernel<<<1, 256, 0, stream>>>(verts, quat, trans, vinit, xy, zcv, nv);
    coef_kernel<<<(nfPad + 255) / 256, 256, 0, stream>>>(xy, zcv, faces, coef, meta, fbox, nf, nfPad);
    cbox_kernel<<<(nChunksPad + 255) / 256, 256, 0, stream>>>(fbox, cbox, nChunksPad);
    {
        dim3 grid(IMG_W / 16, IMG_H / 8);
        raster_kernel<<<grid, 256, 0, stream>>>(coef, meta, cbox, img, acc, nChunksPad);
    }
    edge_kernel<<<(ne + 255) / 256, 256, 0, stream>>>(verts, vinit, edges, deg, nb, ne);
    vertex_kernel<<<(nv + 255) / 256, 256, 0, stream>>>(verts, vinit, deg, nb, acc, nv);
    final_kernel<<<1, 1, 0, stream>>>(acc, out);
}